// SinkhornDistance_35605278884402
// MI455X (gfx1250) — compile-verified
//
#include <hip/hip_runtime.h>

#define EPSR (1.0f / 0.1f)   // 1/eps
#define NB   4
#define P    1024
#define ITERS 50

typedef float v2f __attribute__((ext_vector_type(2)));
typedef float v8f __attribute__((ext_vector_type(8)));

// ---------------------------------------------------------------------------
// K = exp(-C/eps) into the pi region of d_out (reused as K during iterations),
// copy C into the C output region, init scaling vectors a=b=1.
// ---------------------------------------------------------------------------
__global__ void __launch_bounds__(256)
sink_init(const float* __restrict__ C, float* __restrict__ K,
          float* __restrict__ Cout, float* __restrict__ a, float* __restrict__ b) {
    int idx = blockIdx.x * 256 + threadIdx.x;          // exactly NB*P*P threads
    float c = C[idx];
    K[idx] = __expf(-c * EPSR);
    Cout[idx] = c;
    if (idx < NB * P) { a[idx] = 1.0f; b[idx] = 1.0f; }
}

// ---------------------------------------------------------------------------
// Row pass: a = (mu + 1e-8) / (K b).
// One wave handles 16 rows via V_WMMA_F32_16X16X4_F32 accumulation over k.
// A-tile layout (32-bit 16x4): lane L -> row (L&15), k-pair 2*(L>>4).
// B is the b segment broadcast across all 16 N columns, so every column of D
// ends up holding the row sums; lanes 0 and 16 write them out.
// ---------------------------------------------------------------------------
__global__ void __launch_bounds__(256)
sink_row(const float* __restrict__ K, const float* __restrict__ b,
         const float* __restrict__ mu, float* __restrict__ a) {
    __shared__ float bs[P];
    const int batch = (blockIdx.x * 128) >> 10;        // 128 rows/block, batch-aligned
    for (int t = threadIdx.x; t < P; t += 256) bs[t] = b[batch * P + t];
    __syncthreads();

    const int wave = threadIdx.x >> 5;
    const int lane = threadIdx.x & 31;
    const int row0 = blockIdx.x * 128 + wave * 16;
    const int m    = lane & 15;
    const int koff = (lane >> 4) << 1;                 // 0 for lanes 0-15, 2 for 16-31
    const float* Krow = K + (size_t)(row0 + m) * P + koff;

    v8f acc = {0.f, 0.f, 0.f, 0.f, 0.f, 0.f, 0.f, 0.f};
    #pragma unroll 4
    for (int k0 = 0; k0 < P; k0 += 4) {
        v2f av = *(const v2f*)(Krow + k0);             // K[row][k0+koff .. +1], b64 load
        v2f bv;
        bv.x = bs[k0 + koff];
        bv.y = bs[k0 + koff + 1];
        acc = __builtin_amdgcn_wmma_f32_16x16x4_f32(
            /*neg_a=*/false, av, /*neg_b=*/false, bv,
            /*c_mod=*/(short)0, acc, /*reuse_a=*/false, /*reuse_b=*/false);
    }

    // D layout: vgpr i holds M=i (lanes 0-15) and M=i+8 (lanes 16-31), N=lane&15.
    if ((lane & 15) == 0) {                            // lanes 0 and 16
        const int base = row0 + (lane >> 4) * 8;
        #pragma unroll
        for (int i = 0; i < 8; ++i)
            a[base + i] = (mu[base + i] + 1e-8f) / acc[i];
    }
}

// ---------------------------------------------------------------------------
// Column pass: b = (nu + 1e-8) / (K^T a). One column per thread, coalesced
// row-major reads of K; a staged in LDS (broadcast operand).
// ---------------------------------------------------------------------------
__global__ void __launch_bounds__(256)
sink_col(const float* __restrict__ K, const float* __restrict__ a,
         const float* __restrict__ nu, float* __restrict__ b) {
    __shared__ float as[P];
    const int j   = blockIdx.x * 256 + threadIdx.x;    // over NB*P columns
    const int n   = j >> 10;                           // uniform per block (256 | 1024)
    const int col = j & (P - 1);
    for (int t = threadIdx.x; t < P; t += 256) as[t] = a[n * P + t];
    __syncthreads();

    const float* Kp = K + ((size_t)n << 20) + col;
    float s = 0.f;
    #pragma unroll 8
    for (int i = 0; i < P; ++i)
        s = fmaf(as[i], Kp[(size_t)i * P], s);
    b[j] = (nu[j] + 1e-8f) / s;
}

// ---------------------------------------------------------------------------
// pi = a_i * K_ij * b_j written in place over K; per-block partial of
// sum(pi*C) into ws for a deterministic two-stage cost reduction.
// ---------------------------------------------------------------------------
__global__ void __launch_bounds__(256)
sink_pi(float* __restrict__ K, const float* __restrict__ Cin,
        const float* __restrict__ a, const float* __restrict__ b,
        float* __restrict__ partials) {
    __shared__ float red[256];
    const int idx = blockIdx.x * 256 + threadIdx.x;
    const int n = idx >> 20;
    const int i = (idx >> 10) & (P - 1);
    const int j = idx & (P - 1);
    const float pi = a[n * P + i] * K[idx] * b[n * P + j];
    K[idx] = pi;
    red[threadIdx.x] = pi * Cin[idx];
    __syncthreads();
    for (int s = 128; s > 0; s >>= 1) {
        if (threadIdx.x < s) red[threadIdx.x] += red[threadIdx.x + s];
        __syncthreads();
    }
    if (threadIdx.x == 0) partials[blockIdx.x] = red[0];
}

__global__ void __launch_bounds__(256)
sink_cost(const float* __restrict__ partials, float* __restrict__ cost) {
    __shared__ float red[256];
    const float* p = partials + blockIdx.x * 4096;     // 4096 partials per batch
    float s = 0.f;
    for (int t = threadIdx.x; t < 4096; t += 256) s += p[t];
    red[threadIdx.x] = s;
    __syncthreads();
    for (int k = 128; k > 0; k >>= 1) {
        if (threadIdx.x < k) red[threadIdx.x] += red[threadIdx.x + k];
        __syncthreads();
    }
    if (threadIdx.x == 0) cost[blockIdx.x] = red[0];
}

// ---------------------------------------------------------------------------
extern "C" void kernel_launch(void* const* d_in, const int* in_sizes, int n_in,
                              void* d_out, int out_size, void* d_ws, size_t ws_size,
                              hipStream_t stream) {
    const float* mu = (const float*)d_in[0];           // (N,P1)
    const float* nu = (const float*)d_in[1];           // (N,P2)
    const float* C  = (const float*)d_in[2];           // (N,P1,P2)

    float* out  = (float*)d_out;
    float* cost = out;                                 // [0..4)
    float* K    = out + 4;                             // pi region, holds K during iters
    float* Cout = out + 4 + (size_t)NB * P * P;        // C copy region

    float* a        = (float*)d_ws;                    // NB*P
    float* b        = a + NB * P;                      // NB*P
    float* partials = b + NB * P;                      // 16384 block partials

    const int elems = NB * P * P;                      // 4M

    sink_init<<<elems / 256, 256, 0, stream>>>(C, K, Cout, a, b);
    for (int it = 0; it < ITERS; ++it) {
        sink_row<<<(NB * P) / 128, 256, 0, stream>>>(K, b, mu, a);
        sink_col<<<(NB * P) / 256, 256, 0, stream>>>(K, a, nu, b);
    }
    sink_pi<<<elems / 256, 256, 0, stream>>>(K, C, a, b, partials);
    sink_cost<<<NB, 256, 0, stream>>>(partials, cost);
}